// VectorBasis_18588618457096
// MI455X (gfx1250) — compile-verified
//
#include <hip/hip_runtime.h>

typedef float v2f __attribute__((ext_vector_type(2)));
typedef float v8f __attribute__((ext_vector_type(8)));

#define PI_F 3.14159265358979323846f
#define RC_F 5.0f
#define WIDTH_F 0.5f

// Native L2-resident fp32 atomic add: relaxed, agent scope -> global_atomic_add_f32
__device__ __forceinline__ void atomic_add_f32(float* p, float v) {
    __hip_atomic_fetch_add(p, v, __ATOMIC_RELAXED, __HIP_MEMORY_SCOPE_AGENT);
}

// ---------------------------------------------------------------------------
// Kernel 0: zero the spex accumulator (A*96 floats, multiple of 4)
// ---------------------------------------------------------------------------
__global__ void vb_zero_kernel(float4* __restrict__ p, int n4) {
    int i = blockIdx.x * blockDim.x + threadIdx.x;
    int stride = gridDim.x * blockDim.x;
    for (; i < n4; i += stride) p[i] = make_float4(0.f, 0.f, 0.f, 0.f);
}

// ---------------------------------------------------------------------------
// Kernel 1: per-edge expansion + scatter-add into spex (A,3,8,4)
//   153.6M global_atomic_add_f32 into a 38.4MB buffer (fits 192MB L2)
// ---------------------------------------------------------------------------
__global__ void vb_edge_kernel(const float* __restrict__ vecs,   // (E,3)
                               const int*   __restrict__ centers,
                               const int*   __restrict__ neighbors,
                               const int*   __restrict__ species,
                               const float* __restrict__ W_alch, // (4,4)
                               float*       __restrict__ spex,   // (A,96)
                               int E) {
    int e = blockIdx.x * blockDim.x + threadIdx.x;
    if (e >= E) return;

    float vx = vecs[3 * e + 0];
    float vy = vecs[3 * e + 1];
    float vz = vecs[3 * e + 2];
    float d  = sqrtf(vx * vx + vy * vy + vz * vz + 1e-12f);
    float inv_d = __frcp_rn(d);

    // shifted-cosine cutoff (branchless-friendly: compiler -> v_cndmask)
    const float inner = RC_F - WIDTH_F;
    float taper = 0.5f * (__cosf(PI_F * (d - inner) / WIDTH_F) + 1.0f);
    float fc = (d < inner) ? 1.0f : ((d < RC_F) ? taper : 0.0f);

    // radial basis sin(n*pi*d/rc)/d, n=1..8 via Chebyshev recurrence
    float x  = (PI_F / RC_F) * d;
    float s1 = __sinf(x);
    float c1 = __cosf(x);
    float twoc = 2.0f * c1;
    float R[8];
    float sprev = 0.0f, scur = s1;
    #pragma unroll
    for (int n = 0; n < 8; ++n) {
        R[n] = scur * inv_d;
        float snext = twoc * scur - sprev;
        sprev = scur;
        scur  = snext;
    }

    // l=1 harmonics in m-order (-1,0,1) -> (y,z,x)/d, folded with fc
    float ym[3];
    ym[0] = fc * vy * inv_d;
    ym[1] = fc * vz * inv_d;
    ym[2] = fc * vx * inv_d;

    // neighbor alchemical weights (4 floats, 16B aligned row)
    int sp = species[neighbors[e]];
    float4 al = reinterpret_cast<const float4*>(W_alch)[sp];

    float* base = spex + (size_t)centers[e] * 96;
    #pragma unroll
    for (int m = 0; m < 3; ++m) {
        #pragma unroll
        for (int n = 0; n < 8; ++n) {
            float t = ym[m] * R[n];
            float* p = base + m * 32 + n * 4;
            atomic_add_f32(p + 0, t * al.x);
            atomic_add_f32(p + 1, t * al.y);
            atomic_add_f32(p + 2, t * al.z);
            atomic_add_f32(p + 3, t * al.w);
        }
    }
}

// ---------------------------------------------------------------------------
// Kernel 2: per-atom finish via V_WMMA_F32_16X16X4_F32.
//   Rows g = a*3+m (M = 3A = 300000, divisible by 16), K = 32, N = 3 (cols
//   3..15 of B are zero). One wave = one 16-row tile, 8 WMMAs over K.
//   EXEC is all-ones at every WMMA (wave-uniform early-out only).
//   B fragment: clamped-address unconditional loads + branchless zero mask
//   (avoids per-tile EXEC save/restore branches), hoisted out of the K-loop.
// ---------------------------------------------------------------------------
__global__ void vb_finish_kernel(const float* __restrict__ spex,   // (A,96)
                                 const int*   __restrict__ species,
                                 const float* __restrict__ ce,     // (4,32)
                                 const float* __restrict__ Wc,     // (3,32)
                                 float*       __restrict__ out,    // (A,3,3)
                                 int totalRows) {
    int tid  = blockIdx.x * blockDim.x + threadIdx.x;
    int wave = tid >> 5;
    int lane = tid & 31;
    int rowBase = wave * 16;
    if (rowBase >= totalRows) return;   // wave-uniform

    int h  = lane >> 4;   // lane half: selects k pair {2h, 2h+1} of each K=4 step
    int rl = lane & 15;   // A-frag row / B,D column

    // A-fragment row: g = rowBase + rl -> atom a, vector component m
    int g = rowBase + rl;
    int a = g / 3;
    int m = g - 3 * a;
    const float2* x2 = reinterpret_cast<const float2*>(spex + (size_t)a * 96 + m * 32);
    const float2* c2 = reinterpret_cast<const float2*>(ce + species[a] * 32);

    // B-fragment column rl: row rl of W_contract; rows >= 3 are zero.
    // Clamp the address (always valid) and mask the value -> no divergence.
    int   rcl   = (rl < 3) ? rl : 0;
    float bmask = (rl < 3) ? 1.0f : 0.0f;
    const float2* w2 = reinterpret_cast<const float2*>(Wc + rcl * 32);

    v2f bfrag[8];
    #pragma unroll
    for (int t = 0; t < 8; ++t) {
        float2 wv = w2[2 * t + h];     // k = 4t + 2h
        bfrag[t].x = wv.x * bmask;
        bfrag[t].y = wv.y * bmask;
    }

    v8f acc = {};
    #pragma unroll
    for (int t = 0; t < 8; ++t) {
        int p = 2 * t + h;             // float2 index: k = 4t + 2h
        float2 xv = x2[p];
        float2 cv = c2[p];
        v2f afrag;
        afrag.x = xv.x * cv.x;
        afrag.y = xv.y * cv.y;
        acc = __builtin_amdgcn_wmma_f32_16x16x4_f32(
            /*neg_a=*/false, afrag, /*neg_b=*/false, bfrag[t],
            /*c_mod=*/(short)0, acc, /*reuse_a=*/false, /*reuse_b=*/false);
    }

    // D: VGPR j holds row j + 8*h, column rl
    if (rl < 3) {
        #pragma unroll
        for (int j = 0; j < 8; ++j) {
            int grow = rowBase + j + 8 * h;
            int aa = grow / 3;
            int mm = grow - 3 * aa;
            out[(size_t)aa * 9 + mm * 3 + rl] = acc[j];
        }
    }
}

// ---------------------------------------------------------------------------
extern "C" void kernel_launch(void* const* d_in, const int* in_sizes, int n_in,
                              void* d_out, int out_size, void* d_ws, size_t ws_size,
                              hipStream_t stream) {
    const float* vecs      = (const float*)d_in[0];
    const int*   centers   = (const int*)  d_in[1];
    const int*   neighbors = (const int*)  d_in[2];
    const int*   species   = (const int*)  d_in[3];
    // d_in[4] structures, d_in[5] atom_index_in_structure: unused by reference
    const float* W_alch    = (const float*)d_in[6];
    const float* ce        = (const float*)d_in[7];
    const float* Wc        = (const float*)d_in[8];
    float*       out       = (float*)d_out;
    float*       spex      = (float*)d_ws;   // A*96 floats = 38.4 MB

    const int E = in_sizes[1];   // centers count
    const int A = in_sizes[3];   // species count

    // zero accumulator
    int n4 = (A * 96) / 4;
    int zblocks = (n4 + 255) / 256;
    if (zblocks > 4096) zblocks = 4096;
    vb_zero_kernel<<<zblocks, 256, 0, stream>>>((float4*)spex, n4);

    // edge scatter
    vb_edge_kernel<<<(E + 255) / 256, 256, 0, stream>>>(
        vecs, centers, neighbors, species, W_alch, spex, E);

    // WMMA finish
    int totalRows = A * 3;
    int waves   = (totalRows + 15) / 16;
    int threads = waves * 32;
    vb_finish_kernel<<<(threads + 255) / 256, 256, 0, stream>>>(
        spex, species, ce, Wc, out, totalRows);
}